// KGAT_86955907875600
// MI455X (gfx1250) — compile-verified
//
#include <hip/hip_runtime.h>
#include <math.h>

#define N_USERS   50000
#define N_ENT     100000
#define NV        150000      // N_USERS + N_ENT
#define D         64
#define E_UI      1500000

#define NBLK_LOGIT 2048

typedef __attribute__((ext_vector_type(2))) float v2f;
typedef __attribute__((ext_vector_type(8))) float v8f;

// ---------------------------------------------------------------------------
// Kernel 1: zero output accumulator, init scalars (scal[0]=max, scal[1]=Z)
// ---------------------------------------------------------------------------
__global__ void k_init(float* __restrict__ out, float* __restrict__ scal) {
    const size_t n4 = (size_t)NV * D / 4;
    float4 z = make_float4(0.f, 0.f, 0.f, 0.f);
    for (size_t i = (size_t)blockIdx.x * blockDim.x + threadIdx.x; i < n4;
         i += (size_t)gridDim.x * blockDim.x)
        ((float4*)out)[i] = z;
    if (blockIdx.x == 0 && threadIdx.x == 0) {
        scal[0] = -INFINITY;   // global max of logits
        scal[1] = 0.f;         // softmax denominator
    }
}

// ---------------------------------------------------------------------------
// Kernel 2: T = X @ W via V_WMMA_F32_16X16X4_F32 (f32 WMMA, exact precision)
//   X = concat(user_table[user_idx], entity_table[item_idx])  [NV x 64]
//   One wave computes one 16x16 tile of T; K=64 -> 16 chained WMMAs.
//   A 16x4 f32 layout: lanes 0-15 hold M=lane, K={k,k+1}; lanes 16-31 K={k+2,k+3}
//   B 4x16 f32 layout: mirrors A (N=lane&15, same K split across lane halves)
//   C/D: VGPR j -> M = (lane>=16 ? 8 : 0)+j, N = lane&15
// ---------------------------------------------------------------------------
__global__ void k_gemm(const int*   __restrict__ user_idx,
                       const int*   __restrict__ item_idx,
                       const float* __restrict__ userT,
                       const float* __restrict__ entT,
                       const float* __restrict__ W,
                       float*       __restrict__ T) {
    const int wave  = blockIdx.x * (blockDim.x >> 5) + (threadIdx.x >> 5);
    const int lane  = threadIdx.x & 31;
    const int mtile = wave >> 2;          // 9375 M-tiles
    const int ntile = wave & 3;           // 4 N-tiles
    if (mtile >= NV / 16) return;         // wave-uniform exit: EXEC stays all-ones

    const int l16  = lane & 15;
    const int koff = (lane >> 4) * 2;     // K sub-offset per lane half
    const int row  = mtile * 16 + l16;    // A row for this lane
    const int col  = ntile * 16 + l16;    // B/C column for this lane

    // honor the (identity) index gathers from the reference
    const float* xrow;
    if (row < N_USERS) xrow = userT + (size_t)user_idx[row] * D;
    else               xrow = entT  + (size_t)item_idx[row - N_USERS] * D;

    v8f acc = {0.f, 0.f, 0.f, 0.f, 0.f, 0.f, 0.f, 0.f};
#pragma unroll
    for (int k = 0; k < D; k += 4) {
        v2f a, b;
        a.x = xrow[k + koff];
        a.y = xrow[k + koff + 1];
        b.x = W[(size_t)(k + koff)     * D + col];
        b.y = W[(size_t)(k + koff + 1) * D + col];
        acc = __builtin_amdgcn_wmma_f32_16x16x4_f32(
            /*neg_a=*/false, a, /*neg_b=*/false, b,
            /*c_mod=*/(short)0, acc, /*reuse_a=*/false, /*reuse_b=*/false);
    }

    const int mbase = mtile * 16 + (lane >> 4) * 8;
#pragma unroll
    for (int j = 0; j < 8; ++j)
        T[(size_t)(mbase + j) * D + col] = acc[j];
}

// ---------------------------------------------------------------------------
// Kernel 3: per-edge logits = leaky_relu(dot(T[dst], T[src]), 0.2)
//   16 lanes per edge (float4 each = 64 floats), shfl-xor reduce.
//   Also produces per-block max for the global softmax.
// ---------------------------------------------------------------------------
__global__ void k_logits(const int*   __restrict__ edges,
                         const float* __restrict__ T,
                         float*       __restrict__ logits,
                         float*       __restrict__ blockmax) {
    const int lane = threadIdx.x & 31;
    const int half = lane >> 4;
    const int l16  = lane & 15;
    const int wib  = threadIdx.x >> 5;        // wave in block (0..7)
    const int epb  = blockDim.x / 16;         // edges per block-iteration (16)

    float lmax = -INFINITY;
    for (size_t base = (size_t)blockIdx.x * epb; base < E_UI;
         base += (size_t)gridDim.x * epb) {
        size_t e = base + (size_t)wib * 2 + half;
        float val = 0.f;
        int active = (e < E_UI);
        if (active) {
            const int src = edges[e];
            const int dst = edges[E_UI + e];
            const float4 s4 = ((const float4*)(T + (size_t)src * D))[l16];
            const float4 d4 = ((const float4*)(T + (size_t)dst * D))[l16];
            val = s4.x * d4.x + s4.y * d4.y + s4.z * d4.z + s4.w * d4.w;
        }
        // reduce across 16 lanes (xor masks < 16 stay within the half-wave)
        for (int off = 8; off >= 1; off >>= 1)
            val += __shfl_xor(val, off, 32);
        if (l16 == 0 && active) {
            float l = (val > 0.f) ? val : 0.2f * val;   // leaky_relu(0.2)
            logits[e] = l;
            lmax = fmaxf(lmax, l);
        }
    }

    __shared__ float sm[256];
    sm[threadIdx.x] = lmax;
    __syncthreads();
    for (int s = blockDim.x >> 1; s > 0; s >>= 1) {
        if (threadIdx.x < (unsigned)s)
            sm[threadIdx.x] = fmaxf(sm[threadIdx.x], sm[threadIdx.x + s]);
        __syncthreads();
    }
    if (threadIdx.x == 0) blockmax[blockIdx.x] = sm[0];
}

// ---------------------------------------------------------------------------
// Kernel 4: reduce per-block maxima -> scal[0]
// ---------------------------------------------------------------------------
__global__ void k_reduce_max(const float* __restrict__ blockmax, int n,
                             float* __restrict__ scal) {
    __shared__ float sm[1024];
    float m = -INFINITY;
    for (int i = threadIdx.x; i < n; i += blockDim.x)
        m = fmaxf(m, blockmax[i]);
    sm[threadIdx.x] = m;
    __syncthreads();
    for (int s = blockDim.x >> 1; s > 0; s >>= 1) {
        if (threadIdx.x < (unsigned)s)
            sm[threadIdx.x] = fmaxf(sm[threadIdx.x], sm[threadIdx.x + s]);
        __syncthreads();
    }
    if (threadIdx.x == 0) scal[0] = sm[0];
}

// ---------------------------------------------------------------------------
// Kernel 5: Z = sum(exp(logit - m))  -> scal[1]
// ---------------------------------------------------------------------------
__global__ void k_expsum(const float* __restrict__ logits,
                         float* __restrict__ scal) {
    const float m = scal[0];
    float s = 0.f;
    for (size_t i = (size_t)blockIdx.x * blockDim.x + threadIdx.x; i < E_UI;
         i += (size_t)gridDim.x * blockDim.x)
        s += expf(logits[i] - m);

    __shared__ float sm[256];
    sm[threadIdx.x] = s;
    __syncthreads();
    for (int st = blockDim.x >> 1; st > 0; st >>= 1) {
        if (threadIdx.x < (unsigned)st)
            sm[threadIdx.x] += sm[threadIdx.x + st];
        __syncthreads();
    }
    if (threadIdx.x == 0) atomicAdd(&scal[1], sm[0]);
}

// ---------------------------------------------------------------------------
// Kernel 6: scatter msg = score * T[src] into out[dst] (segment_sum)
// ---------------------------------------------------------------------------
__global__ void k_scatter(const int*   __restrict__ edges,
                          const float* __restrict__ T,
                          const float* __restrict__ logits,
                          const float* __restrict__ scal,
                          float*       __restrict__ out) {
    const float m    = scal[0];
    const float invZ = 1.0f / scal[1];
    const int lane = threadIdx.x & 31;
    const int half = lane >> 4;
    const int l16  = lane & 15;
    const int wib  = threadIdx.x >> 5;
    const int epb  = blockDim.x / 16;

    for (size_t base = (size_t)blockIdx.x * epb; base < E_UI;
         base += (size_t)gridDim.x * epb) {
        size_t e = base + (size_t)wib * 2 + half;
        if (e >= E_UI) continue;
        const int src = edges[e];
        const int dst = edges[E_UI + e];
        const float score = expf(logits[e] - m) * invZ;
        const float4 s4 = ((const float4*)(T + (size_t)src * D))[l16];
        float* o = out + (size_t)dst * D + (size_t)l16 * 4;
        atomicAdd(o + 0, score * s4.x);
        atomicAdd(o + 1, score * s4.y);
        atomicAdd(o + 2, score * s4.z);
        atomicAdd(o + 3, score * s4.w);
    }
}

// ---------------------------------------------------------------------------
// Kernel 7: in-place relu on out
// ---------------------------------------------------------------------------
__global__ void k_relu(float* __restrict__ out) {
    const size_t n4 = (size_t)NV * D / 4;
    for (size_t i = (size_t)blockIdx.x * blockDim.x + threadIdx.x; i < n4;
         i += (size_t)gridDim.x * blockDim.x) {
        float4 v = ((float4*)out)[i];
        v.x = fmaxf(v.x, 0.f); v.y = fmaxf(v.y, 0.f);
        v.z = fmaxf(v.z, 0.f); v.w = fmaxf(v.w, 0.f);
        ((float4*)out)[i] = v;
    }
}

// ---------------------------------------------------------------------------
extern "C" void kernel_launch(void* const* d_in, const int* in_sizes, int n_in,
                              void* d_out, int out_size, void* d_ws, size_t ws_size,
                              hipStream_t stream) {
    const int*   user_idx = (const int*)  d_in[0];
    const int*   item_idx = (const int*)  d_in[1];
    const int*   edges_ui = (const int*)  d_in[2];
    // d_in[3] edge_index_kg, d_in[4] edge_type_kg, d_in[8] W_r: dead code in ref
    const float* userT    = (const float*)d_in[5];
    const float* entT     = (const float*)d_in[6];
    const float* W        = (const float*)d_in[7];
    float*       out      = (float*)d_out;

    // workspace layout
    float* T        = (float*)d_ws;                 // NV*D      = 9.6M floats
    float* logits   = T + (size_t)NV * D;           // E_UI      = 1.5M floats
    float* blockmax = logits + E_UI;                // NBLK_LOGIT floats
    float* scal     = blockmax + NBLK_LOGIT;        // 2 floats

    // 1) zero accumulator + scalars
    k_init<<<1024, 256, 0, stream>>>(out, scal);

    // 2) T = X @ W   (37500 wave-tiles, 8 waves/block)
    {
        int waves  = (NV / 16) * 4;
        int blocks = (waves + 7) / 8;
        k_gemm<<<blocks, 256, 0, stream>>>(user_idx, item_idx, userT, entT, W, T);
    }

    // 3) logits + per-block max
    k_logits<<<NBLK_LOGIT, 256, 0, stream>>>(edges_ui, T, logits, blockmax);

    // 4) global max
    k_reduce_max<<<1, 1024, 0, stream>>>(blockmax, NBLK_LOGIT, scal);

    // 5) softmax denominator
    k_expsum<<<1024, 256, 0, stream>>>(logits, scal);

    // 6) weighted scatter-add
    k_scatter<<<4096, 256, 0, stream>>>(edges_ui, T, logits, scal, out);

    // 7) relu
    k_relu<<<2048, 256, 0, stream>>>(out);
}